// SinkhornLoss_2241972929057
// MI455X (gfx1250) — compile-verified
//
#include <hip/hip_runtime.h>
#include <math.h>

// Sinkhorn divergence (geomloss, tensorized) for B=32, N=M=1024, D=8, f32.
// Fully fused: cost matrices never materialized. Per 16x16 (i,j) tile the
// dot-products come from two chained V_WMMA_F32_16X16X4_F32 ops (K=8 = 2*4);
// a streaming logsumexp runs in the WMMA accumulator layout. Q-points and the
// per-column h-vector are staged in LDS once per block (async-to-LDS when the
// toolchain exposes the gfx1250 builtin).

#define B_ 32
#define N_ 1024
#define D_ 8
#define MAX_EPS 32
#define MAX_ITERS 24

typedef __attribute__((ext_vector_type(2))) float v2f;
typedef __attribute__((ext_vector_type(8))) float v8f;
typedef __attribute__((ext_vector_type(4))) int   v4i;

#define LOG2E_ 1.4426950408889634f
#define LN2_   0.6931471805599453f
#define LOGN_  6.9314718055994531f   /* ln(1024) */

// workspace layout (float offsets)
#define WS_X2    0                   /* 32768: ||x_i||^2             */
#define WS_Y2    32768               /* 32768: ||y_j||^2             */
#define WS_POTS  65536               /* 4*32768: f_ba,g_ab,f_aa,g_bb */
#define WS_TMPS  196608              /* 4*32768: tmp softmin outputs */
#define WS_SCHED 327680              /* MAX_EPS eps values + count   */

#if __has_builtin(__builtin_amdgcn_global_load_async_to_lds_b128)
#define HAVE_ASYNC_LDS 1
#else
#define HAVE_ASYNC_LDS 0
#endif

__device__ __forceinline__ void wait_async0() {
#if __has_builtin(__builtin_amdgcn_s_wait_asynccnt)
  __builtin_amdgcn_s_wait_asynccnt(0);
#else
  asm volatile("s_wait_asynccnt 0" ::: "memory");
#endif
}

// ---------------------------------------------------------------------------
// Prep: squared norms + zero-init potentials.
__global__ __launch_bounds__(256) void prep_kernel(const float* __restrict__ x,
                                                   const float* __restrict__ y,
                                                   float* __restrict__ ws) {
  int t = blockIdx.x * 256 + threadIdx.x;      // 0..131071
  ws[WS_POTS + t] = 0.0f;
  if (t < 65536) {
    const float* p = (t < 32768) ? (x + (size_t)t * D_)
                                 : (y + (size_t)(t - 32768) * D_);
    float s = 0.0f;
#pragma unroll
    for (int d = 0; d < D_; ++d) { float v = p[d]; s += v * v; }
    ws[t] = s;                                 // x2 then y2, contiguous
  }
}

// ---------------------------------------------------------------------------
// Diameter + geomloss epsilon schedule (data-dependent length L, on device).
__global__ __launch_bounds__(256) void sched_kernel(const float* __restrict__ x,
                                                    const float* __restrict__ y,
                                                    float* __restrict__ ws) {
  __shared__ float smn[256][8];
  __shared__ float smx[256][8];
  int t = threadIdx.x;
  float mn[8], mx[8];
#pragma unroll
  for (int d = 0; d < 8; ++d) { mn[d] = 3.0e38f; mx[d] = -3.0e38f; }
  for (int p = t; p < 65536; p += 256) {
    const float* base = (p < 32768) ? (x + (size_t)p * D_)
                                    : (y + (size_t)(p - 32768) * D_);
#pragma unroll
    for (int d = 0; d < 8; ++d) {
      float v = base[d];
      mn[d] = fminf(mn[d], v);
      mx[d] = fmaxf(mx[d], v);
    }
  }
#pragma unroll
  for (int d = 0; d < 8; ++d) { smn[t][d] = mn[d]; smx[t][d] = mx[d]; }
  __syncthreads();
  for (int stride = 128; stride > 0; stride >>= 1) {
    if (t < stride) {
#pragma unroll
      for (int d = 0; d < 8; ++d) {
        smn[t][d] = fminf(smn[t][d], smn[t + stride][d]);
        smx[t][d] = fmaxf(smx[t][d], smx[t + stride][d]);
      }
    }
    __syncthreads();
  }
  if (t == 0) {
    float ss = 0.0f;
#pragma unroll
    for (int d = 0; d < 8; ++d) { float r = smx[0][d] - smn[0][d]; ss += r * r; }
    float diam = sqrtf(ss);
    float* sched = ws + WS_SCHED;
    int idx = 0;
    sched[idx++] = diam * diam;                          // d^P
    float e    = 2.0f * (__builtin_amdgcn_logf(diam) * LN2_); // P*ln(d)
    float stop = -2.0f * LN2_;                           // P*ln(blur), blur=0.5
    float step = stop;                                   // P*ln(scaling)
    while (e > stop && idx < MAX_EPS - 1) {              // np.arange semantics
      sched[idx++] = __builtin_amdgcn_exp2f(e * LOG2E_);
      e += step;
    }
    sched[idx++] = 0.25f;                                // blur^P
    int L = idx;
    for (; idx < MAX_EPS; ++idx) sched[idx] = 0.25f;
    sched[MAX_EPS] = (float)L;
  }
}

// ---------------------------------------------------------------------------
// Generic softmin sweep:  out[b,i] = -eps * logsumexp_j( h_j - C(P_i,Q_j)/eps )
// with h_j = -ln(N) + pot_j/eps  (pot ignored in init mode).
// mode: -2 = init (write potentials directly), -1 = final extrapolation
// (eps = blur^2 = 0.25), >=0 = annealing iteration index (no-op if >= L).
// One block = (matrix, batch, 128-row block); one wave = 16 rows.
// Q points (32KB) + precomputed per-column term (4KB) staged in LDS.
__global__ __launch_bounds__(256) void softmin_kernel(const float* __restrict__ x,
                                                      const float* __restrict__ y,
                                                      float* __restrict__ ws,
                                                      int mode) {
  __shared__ float lds_q[N_ * D_];   // 32 KB: Q rows (8 floats each)
  __shared__ float lds_h[N_];        //  4 KB: log2e*h_c - 0.5*c1*|q_c|^2

  const float* sched = ws + WS_SCHED;
  int L = (int)sched[MAX_EPS];
  if (mode >= 0 && mode >= L) return;            // uniform early-out
  float eps;
  if (mode == -2)      eps = sched[0];
  else if (mode == -1) eps = 0.25f;
  else                 eps = sched[mode];
  const float inv_eps = 1.0f / eps;

  int blk    = blockIdx.x;
  int mat    = blk >> 8;        // 0: xy, 1: yx, 2: xx, 3: yy
  int rem    = blk & 255;
  int batch  = rem >> 3;
  int rowblk = rem & 7;

  float* pots = ws + WS_POTS;
  float* tmps = ws + WS_TMPS;
  const float *P, *Q, *p2, *q2, *pot;
  float* out;
  const bool init = (mode == -2);
  switch (mat) {
    case 0:  P = x; Q = y; p2 = ws + WS_X2; q2 = ws + WS_Y2;
             pot = pots + 1 * 32768; out = (init ? pots : tmps) + 0 * 32768; break;
    case 1:  P = y; Q = x; p2 = ws + WS_Y2; q2 = ws + WS_X2;
             pot = pots + 0 * 32768; out = (init ? pots : tmps) + 1 * 32768; break;
    case 2:  P = x; Q = x; p2 = ws + WS_X2; q2 = ws + WS_X2;
             pot = pots + 2 * 32768; out = (init ? pots : tmps) + 2 * 32768; break;
    default: P = y; Q = y; p2 = ws + WS_Y2; q2 = ws + WS_Y2;
             pot = pots + 3 * 32768; out = (init ? pots : tmps) + 3 * 32768; break;
  }
  size_t bo = (size_t)batch * N_;
  P += bo * D_; Q += bo * D_;
  p2 += bo; q2 += bo; pot += bo; out += bo;

  const float c1 = inv_eps * LOG2E_;            // work in log2 domain

  // ---- stage Q into LDS (CDNA5 async-to-LDS path when available) ----------
#if HAVE_ASYNC_LDS
  for (int i = threadIdx.x; i < (N_ * D_) / 4; i += 256) {
    __builtin_amdgcn_global_load_async_to_lds_b128(
        (__attribute__((address_space(1))) v4i*)(Q + (size_t)i * 4),
        (__attribute__((address_space(3))) v4i*)(lds_q + (size_t)i * 4),
        0, 0);
  }
  wait_async0();
#else
  for (int i = threadIdx.x; i < (N_ * D_) / 4; i += 256) {
    float4 v = ((const float4*)Q)[i];
    ((float4*)lds_q)[i] = v;
  }
#endif
  // per-column term: u contribution independent of the row
  for (int c = threadIdx.x; c < N_; c += 256) {
    float potc = init ? 0.0f : pot[c];
    lds_h[c] = LOG2E_ * (potc * inv_eps - LOGN_) - 0.5f * c1 * q2[c];
  }
  __syncthreads();

  int tid  = threadIdx.x;
  int wave = tid >> 5;
  int lane = tid & 31;
  int lh   = lane >> 4;         // lane half (0/1)
  int ll   = lane & 15;
  int rbase = rowblk * 128 + wave * 16;

  // A fragment: 16 rows x K=8 (two K=4 halves).  ISA layout: lane = M,
  // VGPR0/1 hold K={0,1} (lanes 0-15) / K={2,3} (lanes 16-31); +4 for hi.
  const float* prow = P + (size_t)(rbase + ll) * D_;
  const int k0 = lh * 2;
  v2f aLo = { prow[k0],     prow[k0 + 1] };
  v2f aHi = { prow[k0 + 4], prow[k0 + 5] };

  float x2h[8];
#pragma unroll
  for (int v = 0; v < 8; ++v)                   // row of acc VGPR v for this lane
    x2h[v] = 0.5f * c1 * p2[rbase + v + 8 * lh];

  float m[8], s[8];
#pragma unroll
  for (int v = 0; v < 8; ++v) { m[v] = -3.0e38f; s[v] = 0.0f; }

  for (int ct = 0; ct < 64; ++ct) {             // 64 column tiles of 16
    int c = ct * 16 + ll;                       // this lane's column
    const float* qr = lds_q + (size_t)c * D_;   // LDS-resident Q row
    v2f bLo = { qr[k0],     qr[k0 + 1] };       // B: lane = N, K per half
    v2f bHi = { qr[k0 + 4], qr[k0 + 5] };
    float tc = lds_h[c];
    v8f acc = {0.f,0.f,0.f,0.f,0.f,0.f,0.f,0.f};
    acc = __builtin_amdgcn_wmma_f32_16x16x4_f32(false, aLo, false, bLo,
                                                (short)0, acc, false, false);
    acc = __builtin_amdgcn_wmma_f32_16x16x4_f32(false, aHi, false, bHi,
                                                (short)0, acc, false, false);
#pragma unroll
    for (int v = 0; v < 8; ++v) {
      float u  = fmaf(acc[v], c1, tc - x2h[v]); // log2e*(h_c - C/eps)
      float nm = fmaxf(m[v], u);
      s[v] = s[v] * __builtin_amdgcn_exp2f(m[v] - nm)
                  + __builtin_amdgcn_exp2f(u - nm);
      m[v] = nm;
    }
  }

  // Row reduction across the 16 lanes of each half (butterfly).
#pragma unroll
  for (int off = 1; off < 16; off <<= 1) {
#pragma unroll
    for (int v = 0; v < 8; ++v) {
      float mo = __shfl_xor(m[v], off, 32);
      float so = __shfl_xor(s[v], off, 32);
      float nm = fmaxf(m[v], mo);
      s[v] = s[v] * __builtin_amdgcn_exp2f(m[v] - nm)
           + so   * __builtin_amdgcn_exp2f(mo  - nm);
      m[v] = nm;
    }
  }
  if (ll == 0) {
#pragma unroll
    for (int v = 0; v < 8; ++v) {
      int row = rbase + v + 8 * lh;
      out[row] = -eps * LN2_ * (m[v] + __builtin_amdgcn_logf(s[v]));
    }
  }
}

// ---------------------------------------------------------------------------
// Averaged Sinkhorn update: pot = 0.5*(pot + tmp), all four arrays at once.
__global__ __launch_bounds__(256) void update_kernel(float* __restrict__ ws, int iter) {
  const float* sched = ws + WS_SCHED;
  int L = (int)sched[MAX_EPS];
  if (iter >= L) return;
  int t = blockIdx.x * 256 + threadIdx.x;      // 0..131071
  float* pots = ws + WS_POTS;
  const float* tmps = ws + WS_TMPS;
  pots[t] = 0.5f * (pots[t] + tmps[t]);
}

// ---------------------------------------------------------------------------
// loss = mean_b [ <a, f_ba - f_aa> + <b, g_ab - g_bb> ],  a=b=1/1024 uniform.
__global__ __launch_bounds__(256) void loss_kernel(const float* __restrict__ ws,
                                                   float* __restrict__ out) {
  __shared__ float red[256];
  const float* tmps = ws + WS_TMPS;            // final extrapolated potentials
  int t = threadIdx.x;
  float acc = 0.0f;
  for (int i = t; i < 32768; i += 256) {
    acc += (tmps[i]             - tmps[2 * 32768 + i])    // f_ba - f_aa
         + (tmps[32768 + i]     - tmps[3 * 32768 + i]);   // g_ab - g_bb
  }
  red[t] = acc;
  __syncthreads();
  for (int st = 128; st > 0; st >>= 1) {
    if (t < st) red[t] += red[t + st];
    __syncthreads();
  }
  if (t == 0) out[0] = red[0] * (1.0f / 32768.0f);        // /(N) then /B
}

// ---------------------------------------------------------------------------
extern "C" void kernel_launch(void* const* d_in, const int* in_sizes, int n_in,
                              void* d_out, int out_size, void* d_ws, size_t ws_size,
                              hipStream_t stream) {
  (void)in_sizes; (void)n_in; (void)out_size; (void)ws_size;
  const float* x = (const float*)d_in[0];   // predicted_set [32,1024,8]
  const float* y = (const float*)d_in[1];   // target_set    [32,1024,8]
  float* ws  = (float*)d_ws;
  float* out = (float*)d_out;

  prep_kernel<<<512, 256, 0, stream>>>(x, y, ws);
  sched_kernel<<<1, 256, 0, stream>>>(x, y, ws);
  softmin_kernel<<<1024, 256, 0, stream>>>(x, y, ws, -2);     // init potentials
  for (int it = 0; it < MAX_ITERS; ++it) {                    // eps annealing
    softmin_kernel<<<1024, 256, 0, stream>>>(x, y, ws, it);
    update_kernel<<<512, 256, 0, stream>>>(ws, it);
  }
  softmin_kernel<<<1024, 256, 0, stream>>>(x, y, ws, -1);     // final extrapolation
  loss_kernel<<<1, 256, 0, stream>>>(ws, out);
}